// DeformableEncoderLayer_39067022524862
// MI455X (gfx1250) — compile-verified
//
#include <hip/hip_runtime.h>
#include <hip/hip_bf16.h>
#include <stdint.h>

#define D_MODEL  256
#define D_FFN    1024
#define N_HEADS  8
#define NQ_TOT   21760
#define BATCH    2
#define M_TOT    (BATCH * NQ_TOT)   /* 43520 */
#define ROW_TILES (M_TOT / 16)      /* 2720  */

typedef float v2f __attribute__((ext_vector_type(2)));
typedef float v8f __attribute__((ext_vector_type(8)));

__constant__ int c_dimW[4]   = {128, 64, 32, 16};
__constant__ int c_dimH[4]   = {128, 64, 32, 16};
__constant__ int c_lstart[4] = {0, 16384, 20480, 21504};

// One K=4 step of a 16x16 f32 WMMA tile.
//   sAm : LDS A tile, 16 rows x LD cols (row-major, padded stride LD)
//   bptr: &B[(size_t)k*ldb + col] where col = n0 + lrow  (B row-major KxN)
// A frag (ISA 7.12.2, 32-bit 16x4): lanes 0-15 -> K k+0,k+1 ; lanes 16-31 -> K k+2,k+3
// C/D layout: VGPR r holds row (r + 8*half), col = lane&15
template <int LD>
static __device__ __forceinline__ v8f gemm_step(const float (*sAm)[LD], int k,
                                                const float* __restrict__ bptr,
                                                int ldb, v8f acc, int half, int lrow) {
#if __has_builtin(__builtin_amdgcn_wmma_f32_16x16x4_f32)
  v2f a, b;
  a.x = sAm[lrow][k + 2 * half + 0];
  a.y = sAm[lrow][k + 2 * half + 1];
  b.x = bptr[(size_t)(2 * half + 0) * ldb];
  b.y = bptr[(size_t)(2 * half + 1) * ldb];
  return __builtin_amdgcn_wmma_f32_16x16x4_f32(false, a, false, b, (short)0, acc,
                                               false, false);
#else
#pragma unroll
  for (int kk = 0; kk < 4; ++kk) {
    const float bv = bptr[(size_t)kk * ldb];
#pragma unroll
    for (int r = 0; r < 8; ++r) acc[r] += sAm[r + 8 * half][k + kk] * bv;
  }
  return acc;
#endif
}

// ---------------------------------------------------------------- K1: value
__global__ __launch_bounds__(256) void k_value(const float* __restrict__ src,
                                               const unsigned char* __restrict__ mask,
                                               const float* __restrict__ wv,
                                               const float* __restrict__ bv,
                                               float* __restrict__ value) {
  __shared__ float sA[16][260];
  const int rowbase = blockIdx.x * 16;
  for (int i = threadIdx.x; i < 16 * 256; i += 256) {
    const int r = i >> 8, c = i & 255;
    sA[r][c] = src[(size_t)(rowbase + r) * 256 + c];
  }
  __syncthreads();
  const int lane = threadIdx.x & 31, wave = threadIdx.x >> 5;
  const int half = lane >> 4, lrow = lane & 15;
  for (int t = 0; t < 2; ++t) {
    const int n0 = (wave * 2 + t) * 16;
    const int col = n0 + lrow;
    v8f acc = {};
    for (int k = 0; k < 256; k += 4)
      acc = gemm_step<260>(sA, k, wv + (size_t)k * 256 + col, 256, acc, half, lrow);
#pragma unroll
    for (int r = 0; r < 8; ++r) {
      const int row = rowbase + r + 8 * half;
      const float m = mask[row] ? 0.f : 1.f;
      value[(size_t)row * 256 + col] = (acc[r] + bv[col]) * m;
    }
  }
}

// ------------------------------------------- K2: sampling locations + softmax
__global__ __launch_bounds__(256) void k_loc_attn(const float* __restrict__ src,
                                                  const float* __restrict__ pos,
                                                  const float* __restrict__ refp,
                                                  const float* __restrict__ wsamp,
                                                  const float* __restrict__ bsamp,
                                                  const float* __restrict__ wattn,
                                                  const float* __restrict__ battn,
                                                  float* __restrict__ loc,
                                                  float* __restrict__ attnw) {
  __shared__ float sA[16][260];
  const int rowbase = blockIdx.x * 16;
  for (int i = threadIdx.x; i < 16 * 256; i += 256) {
    const int r = i >> 8, c = i & 255;
    const size_t g = (size_t)(rowbase + r) * 256 + c;
    sA[r][c] = src[g] + pos[g];
  }
  __syncthreads();
  const int lane = threadIdx.x & 31, wave = threadIdx.x >> 5;
  const int half = lane >> 4, lrow = lane & 15;

  // sampling offsets -> absolute locations (N = 256 = H*L*P*2)
  for (int t = 0; t < 2; ++t) {
    const int n0 = (wave * 2 + t) * 16;
    const int col = n0 + lrow;
    v8f acc = {};
    for (int k = 0; k < 256; k += 4)
      acc = gemm_step<260>(sA, k, wsamp + (size_t)k * 256 + col, 256, acc, half, lrow);
    const int l = (col >> 3) & 3;      // col = h*32 + l*8 + p*2 + xy
    const int xy = col & 1;
    const float norm = (float)(xy == 0 ? c_dimW[l] : c_dimH[l]);
    const float bias = bsamp[col];
#pragma unroll
    for (int r = 0; r < 8; ++r) {
      const int row = rowbase + r + 8 * half;
      const int bb = row / NQ_TOT, q = row % NQ_TOT;
      const float ref = refp[((((size_t)bb * NQ_TOT + q) * 4) + l) * 2 + xy];
      loc[(size_t)row * 256 + col] = ref + (acc[r] + bias) / norm;
    }
  }

  // attention logits (N = 128); tile n == head h == wave; softmax over the
  // 16 (L*P) columns of a tile == reduction across a 16-lane half-wave.
  {
    const int col = wave * 16 + lrow;
    v8f acc = {};
    for (int k = 0; k < 256; k += 4)
      acc = gemm_step<260>(sA, k, wattn + (size_t)k * 128 + col, 128, acc, half, lrow);
    const float bias = battn[col];
#pragma unroll
    for (int r = 0; r < 8; ++r) {
      const int row = rowbase + r + 8 * half;
      float v = acc[r] + bias;
      float mx = v;
      for (int s = 1; s < 16; s <<= 1) mx = fmaxf(mx, __shfl_xor(mx, s, 32));
      const float e = __expf(v - mx);
      float sm = e;
      for (int s = 1; s < 16; s <<= 1) sm += __shfl_xor(sm, s, 32);
      attnw[(size_t)row * 128 + col] = e / sm;
    }
  }
}

// -------------------------------------------------- K3: bilinear gather core
__global__ __launch_bounds__(256) void k_sample(const float* __restrict__ value,
                                                const float* __restrict__ loc,
                                                const float* __restrict__ attnw,
                                                float* __restrict__ outp) {
  const int tid = blockIdx.x * 256 + threadIdx.x;  // (m, h)
  const int m = tid >> 3, h = tid & 7;
  const int b = m / NQ_TOT;
  float acc[32];
#pragma unroll
  for (int d = 0; d < 32; ++d) acc[d] = 0.f;
  const float* locm = loc + (size_t)m * 256 + h * 32;
  const float* am = attnw + (size_t)m * 128 + h * 16;
  for (int l = 0; l < 4; ++l) {
    const int W = c_dimW[l], H = c_dimH[l];
    const float* vlev = value + ((size_t)b * NQ_TOT + c_lstart[l]) * 256 + h * 32;
    for (int p = 0; p < 4; ++p) {
      const float lx = locm[l * 8 + p * 2 + 0];
      const float ly = locm[l * 8 + p * 2 + 1];
      const float a = am[l * 4 + p];
      const float x = lx * (float)W - 0.5f;
      const float y = ly * (float)H - 0.5f;
      const float x0f = floorf(x), y0f = floorf(y);
      const int x0 = (int)x0f, y0 = (int)y0f;
      const float wx = x - x0f, wy = y - y0f;
      const int xs[4] = {x0, x0 + 1, x0, x0 + 1};
      const int ys[4] = {y0, y0, y0 + 1, y0 + 1};
      const float wt[4] = {a * (1.f - wy) * (1.f - wx), a * (1.f - wy) * wx,
                           a * wy * (1.f - wx), a * wy * wx};
#pragma unroll
      for (int tc = 0; tc < 4; ++tc) {
        const int xi = xs[tc], yi = ys[tc];
        if (xi >= 0 && xi < W && yi >= 0 && yi < H) {
          const float* vp = vlev + (size_t)(yi * W + xi) * 256;
          const float w = wt[tc];
#pragma unroll
          for (int d = 0; d < 32; d += 4) {
            const float4 vv = *(const float4*)(vp + d);
            acc[d + 0] += w * vv.x;
            acc[d + 1] += w * vv.y;
            acc[d + 2] += w * vv.z;
            acc[d + 3] += w * vv.w;
          }
        }
      }
    }
  }
  float* op = outp + (size_t)m * 256 + h * 32;
#pragma unroll
  for (int d = 0; d < 32; d += 4) {
    float4 vv = {acc[d], acc[d + 1], acc[d + 2], acc[d + 3]};
    *(float4*)(op + d) = vv;
  }
}

// -------------------------------------- K4: out-proj + residual + LayerNorm1
__global__ __launch_bounds__(256) void k_outln(const float* __restrict__ samp,
                                               const float* __restrict__ src,
                                               const float* __restrict__ wo,
                                               const float* __restrict__ bo,
                                               const float* __restrict__ g,
                                               const float* __restrict__ bt,
                                               float* __restrict__ x1) {
  __shared__ float sA[16][260];
  __shared__ float sO[16][260];
  __shared__ float redS[16][16];
  __shared__ float redQ[16][16];
  const int rowbase = blockIdx.x * 16;
  for (int i = threadIdx.x; i < 16 * 256; i += 256) {
    const int r = i >> 8, c = i & 255;
    sA[r][c] = samp[(size_t)(rowbase + r) * 256 + c];
  }
  __syncthreads();
  const int lane = threadIdx.x & 31, wave = threadIdx.x >> 5;
  const int half = lane >> 4, lrow = lane & 15;
  for (int t = 0; t < 2; ++t) {
    const int n0 = (wave * 2 + t) * 16;
    const int col = n0 + lrow;
    v8f acc = {};
    for (int k = 0; k < 256; k += 4)
      acc = gemm_step<260>(sA, k, wo + (size_t)k * 256 + col, 256, acc, half, lrow);
#pragma unroll
    for (int r = 0; r < 8; ++r) {
      const int rl = r + 8 * half;
      sO[rl][col] = acc[r] + bo[col] + src[(size_t)(rowbase + rl) * 256 + col];
    }
  }
  __syncthreads();
  const int r = threadIdx.x >> 4, j = threadIdx.x & 15;
  float s = 0.f, q2 = 0.f;
#pragma unroll
  for (int c = 0; c < 16; ++c) {
    const float v = sO[r][j * 16 + c];
    s += v;
    q2 += v * v;
  }
  redS[r][j] = s;
  redQ[r][j] = q2;
  __syncthreads();
  float mean = 0.f, msq = 0.f;
#pragma unroll
  for (int jj = 0; jj < 16; ++jj) {
    mean += redS[r][jj];
    msq += redQ[r][jj];
  }
  mean *= (1.f / 256.f);
  msq *= (1.f / 256.f);
  const float inv = rsqrtf(msq - mean * mean + 1e-5f);
#pragma unroll
  for (int c = 0; c < 16; ++c) {
    const int col = j * 16 + c;
    x1[(size_t)(rowbase + r) * 256 + col] = (sO[r][col] - mean) * inv * g[col] + bt[col];
  }
}

// --------------------------- K5: fused FFN (hidden in LDS) + residual + LN2
__global__ __launch_bounds__(256) void k_ffn(const float* __restrict__ x1,
                                             const float* __restrict__ w1,
                                             const float* __restrict__ b1,
                                             const float* __restrict__ w2,
                                             const float* __restrict__ b2,
                                             const float* __restrict__ g,
                                             const float* __restrict__ bt,
                                             float* __restrict__ outp) {
  __shared__ float smem[16 * 260 + 16 * 516 + 2 * 16 * 16];
  float(*sA)[260] = (float(*)[260])smem;              // x tile (kept: residual)
  float(*sH)[516] = (float(*)[516])(smem + 16 * 260); // hidden chunk (512 cols)
  float(*sO)[516] = sH;                               // output tile aliases sH
  float(*redS)[16] = (float(*)[16])(smem + 16 * 260 + 16 * 516);
  float(*redQ)[16] = redS + 16;
  const int rowbase = blockIdx.x * 16;
  for (int i = threadIdx.x; i < 16 * 256; i += 256) {
    const int r = i >> 8, c = i & 255;
    sA[r][c] = x1[(size_t)(rowbase + r) * 256 + c];
  }
  __syncthreads();
  const int lane = threadIdx.x & 31, wave = threadIdx.x >> 5;
  const int half = lane >> 4, lrow = lane & 15;
  v8f acc2[2] = {};
  for (int chunk = 0; chunk < 2; ++chunk) {
    const int f0 = chunk * 512;
    for (int t = 0; t < 4; ++t) {  // hidden = relu(x @ w1 + b1), 512-col chunk
      const int hn = (wave * 4 + t) * 16;
      const int fcol = f0 + hn + lrow;
      v8f acc = {};
      for (int k = 0; k < 256; k += 4)
        acc = gemm_step<260>(sA, k, w1 + (size_t)k * 1024 + fcol, 1024, acc, half, lrow);
#pragma unroll
      for (int r = 0; r < 8; ++r)
        sH[r + 8 * half][hn + lrow] = fmaxf(acc[r] + b1[fcol], 0.f);
    }
    __syncthreads();
    for (int t = 0; t < 2; ++t) {  // accumulate hidden @ w2 into registers
      const int n0 = (wave * 2 + t) * 16;
      const int col = n0 + lrow;
      for (int k = 0; k < 512; k += 4)
        acc2[t] = gemm_step<516>(sH, k, w2 + (size_t)(f0 + k) * 256 + col, 256,
                                 acc2[t], half, lrow);
    }
    __syncthreads();
  }
  for (int t = 0; t < 2; ++t) {  // + b2 + residual(x1)
    const int col = (wave * 2 + t) * 16 + lrow;
#pragma unroll
    for (int r = 0; r < 8; ++r) {
      const int rl = r + 8 * half;
      sO[rl][col] = acc2[t][r] + b2[col] + sA[rl][col];
    }
  }
  __syncthreads();
  const int r = threadIdx.x >> 4, j = threadIdx.x & 15;
  float s = 0.f, q2 = 0.f;
#pragma unroll
  for (int c = 0; c < 16; ++c) {
    const float v = sO[r][j * 16 + c];
    s += v;
    q2 += v * v;
  }
  redS[r][j] = s;
  redQ[r][j] = q2;
  __syncthreads();
  float mean = 0.f, msq = 0.f;
#pragma unroll
  for (int jj = 0; jj < 16; ++jj) {
    mean += redS[r][jj];
    msq += redQ[r][jj];
  }
  mean *= (1.f / 256.f);
  msq *= (1.f / 256.f);
  const float inv = rsqrtf(msq - mean * mean + 1e-5f);
#pragma unroll
  for (int c = 0; c < 16; ++c) {
    const int col = j * 16 + c;
    outp[(size_t)(rowbase + r) * 256 + col] = (sO[r][col] - mean) * inv * g[col] + bt[col];
  }
}

extern "C" void kernel_launch(void* const* d_in, const int* in_sizes, int n_in,
                              void* d_out, int out_size, void* d_ws, size_t ws_size,
                              hipStream_t stream) {
  (void)in_sizes; (void)n_in; (void)out_size; (void)ws_size;
  const float* src  = (const float*)d_in[0];
  const float* pos  = (const float*)d_in[1];
  const float* refp = (const float*)d_in[2];
  // d_in[3] spatial_shapes, d_in[4] level_start_index: compile-time constants
  const unsigned char* mask = (const unsigned char*)d_in[5];
  const float* w_samp = (const float*)d_in[6];
  const float* b_samp = (const float*)d_in[7];
  const float* w_attn = (const float*)d_in[8];
  const float* b_attn = (const float*)d_in[9];
  const float* w_val  = (const float*)d_in[10];
  const float* b_val  = (const float*)d_in[11];
  const float* w_out  = (const float*)d_in[12];
  const float* b_out  = (const float*)d_in[13];
  const float* ln1g   = (const float*)d_in[14];
  const float* ln1b   = (const float*)d_in[15];
  const float* w1     = (const float*)d_in[16];
  const float* b1     = (const float*)d_in[17];
  const float* w2     = (const float*)d_in[18];
  const float* b2     = (const float*)d_in[19];
  const float* ln2g   = (const float*)d_in[20];
  const float* ln2b   = (const float*)d_in[21];
  float* out = (float*)d_out;

  float* ws      = (float*)d_ws;
  float* d_value = ws;                             // M*256
  float* d_loc   = d_value + (size_t)M_TOT * 256;  // M*256
  float* d_attn  = d_loc + (size_t)M_TOT * 256;    // M*128
  float* d_samp  = d_attn + (size_t)M_TOT * 128;   // M*256
  float* d_x1    = d_samp + (size_t)M_TOT * 256;   // M*256

  k_value<<<ROW_TILES, 256, 0, stream>>>(src, mask, w_val, b_val, d_value);
  k_loc_attn<<<ROW_TILES, 256, 0, stream>>>(src, pos, refp, w_samp, b_samp, w_attn,
                                            b_attn, d_loc, d_attn);
  k_sample<<<(M_TOT * N_HEADS) / 256, 256, 0, stream>>>(d_value, d_loc, d_attn, d_samp);
  k_outln<<<ROW_TILES, 256, 0, stream>>>(d_samp, src, w_out, b_out, ln1g, ln1b, d_x1);
  k_ffn<<<ROW_TILES, 256, 0, stream>>>(d_x1, w1, b1, w2, b2, ln2g, ln2b, out);
}